// FullSSD300_58136677318682
// MI455X (gfx1250) — compile-verified
//
#include <hip/hip_runtime.h>
#include <stdint.h>

#define B_IMG 32
#define N_BOX 8732
#define N_CLS 81
#define BN (B_IMG * N_BOX)        /* 279424 rows      */
#define NC (N_BOX * N_CLS)        /* 707292           */
#define K_TOP 1024
#define DET_THR 0.4f
#define IOU_THR 0.7f
#define CAND_CAP (2 * BN)         /* <=2 candidates per row, cannot overflow */

typedef unsigned long long u64;
typedef unsigned int u32;
typedef float v2f __attribute__((ext_vector_type(2)));
typedef float v8f __attribute__((ext_vector_type(8)));

/* ---------------- workspace layout (bytes) ---------------- */
#define WS_COUNTER   0
#define WS_CAND      64                                 /* u64[CAND_CAP]   */
#define WS_L1OUT     (WS_CAND + (size_t)CAND_CAP * 8)   /* u64[64*1024]    */
#define WS_TOPK      (WS_L1OUT + 64 * 1024 * 8)         /* u64[1024]       */
#define WS_SB        (WS_TOPK + 1024 * 8)               /* f32[1024*4] shifted boxes */
#define WS_OBOX      (WS_SB + 4096 * 4)                 /* f32[1024*4] original boxes */
#define WS_AREA      (WS_OBOX + 4096 * 4)               /* f32[1024]       */
#define WS_SCORE     (WS_AREA + 1024 * 4)               /* f32[1024]       */
#define WS_CLS       (WS_SCORE + 1024 * 4)              /* i32[1024]       */
#define WS_KEEPINIT  (WS_CLS + 1024 * 4)                /* u32[32]         */
#define WS_SUP       (WS_KEEPINIT + 128)                /* u16[1024*64] = 128KB */

/* ============ Kernel 1: streaming softmax + candidate extraction ============
 * Only the top-2 logits of a row can have softmax prob > 0.4 (probs sum to 1),
 * so a single pass with online (max, second-max, exp-sum) suffices.
 * Memory-bound: one read of 90.5MB @ 23.3 TB/s. Loads are lane-coalesced in n.
 */
__global__ void k_softmax_cand(const float* __restrict__ labels,
                               u32* __restrict__ counter, u64* __restrict__ cand) {
  int gid = blockIdx.x * blockDim.x + threadIdx.x;
  if (gid >= BN) return;
  int b = gid / N_BOX;
  int n = gid - b * N_BOX;
  const float* p = labels + (size_t)b * N_CLS * N_BOX + n;
  float m1 = -__builtin_huge_valf(), m2 = -__builtin_huge_valf();
  int c1 = -1, c2 = -1;
  float S = 0.0f;
  for (int c = 0; c < N_CLS; ++c) {
    float x = p[(size_t)c * N_BOX];
    if (x > m1) {
      S = S * expf(m1 - x) + 1.0f;       /* rescale running sum to new max */
      m2 = m1; c2 = c1; m1 = x; c1 = c;
    } else {
      S += expf(x - m1);
      if (x > m2) { m2 = x; c2 = c; }
    }
  }
  float p1 = 1.0f / S;
  float p2 = expf(m2 - m1) / S;
  /* key = score_bits<<32 | ~idx  -> desc sort == (score desc, idx asc) */
  if (c1 > 0 && p1 > DET_THR) {
    u32 slot = atomicAdd(counter, 1u);
    u32 idx = (u32)gid * N_CLS + (u32)c1;
    cand[slot] = ((u64)__float_as_uint(p1) << 32) | (u32)(~idx);
  }
  if (c2 > 0 && p2 > DET_THR) {
    u32 slot = atomicAdd(counter, 1u);
    u32 idx = (u32)gid * N_CLS + (u32)c2;
    cand[slot] = ((u64)__float_as_uint(p2) << 32) | (u32)(~idx);
  }
}

/* ============ Bitonic top-1024 machinery (exact) ============ */
__device__ __forceinline__ void chunk_to_lds_async(u64* lbuf, const u64* gsrc) {
  /* CDNA5 async global->LDS copy; flat LDS addr low 32 bits == LDS offset. */
  unsigned lds_off = (unsigned)(u64)(uintptr_t)&lbuf[threadIdx.x];
  u64 gaddr = (u64)(uintptr_t)(gsrc + threadIdx.x);
  asm volatile("global_load_async_to_lds_b64 %0, %1, off"
               :: "v"(lds_off), "v"(gaddr) : "memory");
  asm volatile("s_wait_asynccnt 0" ::: "memory");
}

__device__ void bitonic_sort_1024(u64* s, bool asc) {
  int tid = threadIdx.x;
  for (int k = 2; k <= 1024; k <<= 1)
    for (int j = k >> 1; j > 0; j >>= 1) {
      __syncthreads();
      int ixj = tid ^ j;
      if (ixj > tid) {
        bool up = (((tid & k) == 0) == asc);
        u64 a = s[tid], b = s[ixj];
        bool sw = up ? (a > b) : (a < b);
        if (sw) { s[tid] = b; s[ixj] = a; }
      }
    }
  __syncthreads();
}

__device__ void bitonic_merge_desc_1024(u64* s) {
  int tid = threadIdx.x;
  for (int j = 512; j > 0; j >>= 1) {
    __syncthreads();
    int ixj = tid ^ j;
    if (ixj > tid) {
      u64 a = s[tid], b = s[ixj];
      if (a < b) { s[tid] = b; s[ixj] = a; }
    }
  }
  __syncthreads();
}

/* cur (desc) ⊕ asc-sorted chunk -> elementwise max is bitonic -> merge desc. */
__device__ void topk_stream(const u64* src, u32 start, u32 end,
                            u64* cur, u64* buf, u64* out) {
  int tid = threadIdx.x;
  bool first = true;
  for (u32 pos = start; pos < end; pos += 1024u) {
    u64* dst = first ? cur : buf;
    if (pos + 1024u <= end) {
      chunk_to_lds_async(dst, src + pos);
    } else {
      dst[tid] = (pos + (u32)tid < end) ? src[pos + tid] : 0ull;
    }
    __syncthreads();
    if (first) {
      bitonic_sort_1024(cur, false);            /* descending */
      first = false;
    } else {
      bitonic_sort_1024(buf, true);             /* ascending  */
      u64 m = cur[tid] > buf[tid] ? cur[tid] : buf[tid];
      __syncthreads();
      cur[tid] = m;
      bitonic_merge_desc_1024(cur);
    }
  }
  out[tid] = first ? 0ull : cur[tid];
}

__global__ void __launch_bounds__(1024)
k_topk_l1(const u32* __restrict__ counter, const u64* __restrict__ cand,
          u64* __restrict__ l1out) {
  __shared__ u64 cur[1024];
  __shared__ u64 buf[1024];
  u32 M = *counter;
  if (M > (u32)CAND_CAP) M = CAND_CAP;
  u32 per = (M + 63u) >> 6;
  u32 start = blockIdx.x * per;
  u32 end = start + per;
  if (end > M) end = M;
  if (start > M) start = M;
  topk_stream(cand, start, end, cur, buf, l1out + (size_t)blockIdx.x * 1024);
}

__global__ void __launch_bounds__(1024)
k_topk_l2(const u64* __restrict__ l1out, u64* __restrict__ topk) {
  __shared__ u64 cur[1024];
  __shared__ u64 buf[1024];
  topk_stream(l1out, 0u, 64u * 1024u, cur, buf, topk);
}

/* ============ Kernel: decode top-K, boxes, max, offset, areas ============ */
__global__ void __launch_bounds__(1024)
k_decode(const float* __restrict__ locs, const u64* __restrict__ topk,
         float* __restrict__ sb, float* __restrict__ obox, float* __restrict__ area,
         float* __restrict__ score_o, int* __restrict__ cls_o,
         u32* __restrict__ keepInit) {
  __shared__ float red[1024];
  __shared__ u32 fmap[64];        /* 2048-bit candidate bitmap for fillers */
  __shared__ u32 fillIdx[1024];
  __shared__ int cnt;
  int tid = threadIdx.x;
  u64 key = topk[tid];
  bool real = (key != 0ull);
  if (tid == 0) cnt = 0;
  __syncthreads();
  if (real) atomicAdd(&cnt, 1);
  __syncthreads();
  int M1 = cnt;                   /* keys sorted desc -> real == (tid < M1) */
  if (M1 < 1024) {
    /* fillers: smallest flat indices not in candidate set (score == -1)   */
    if (tid < 64) fmap[tid] = 0u;
    __syncthreads();
    if (real) {
      u32 idx = ~(u32)key;
      if (idx < 2048u) atomicOr(&fmap[idx >> 5], 1u << (idx & 31));
    }
    __syncthreads();
    if (tid == 0) {
      int p = M1;
      for (u32 c = 0; c < 2048u && p < 1024; ++c)
        if (!((fmap[c >> 5] >> (c & 31)) & 1u)) fillIdx[p++] = c;
    }
    __syncthreads();
  }
  float score; u32 idx;
  if (real) { score = __uint_as_float((u32)(key >> 32)); idx = ~(u32)key; }
  else      { score = -1.0f; idx = fillIdx[tid]; }

  int cls = (int)(idx % (u32)N_CLS);
  u32 boxr = idx / (u32)N_CLS;
  int img = (int)(idx / (u32)NC) + 1;
  int b = (int)(boxr / (u32)N_BOX);
  int n = (int)(boxr - (u32)b * (u32)N_BOX);
  const float* lp = locs + (size_t)b * 4 * N_BOX + n;   /* locs is (B,4,N) */
  float x1 = lp[0], y1 = lp[(size_t)N_BOX], x2 = lp[(size_t)2 * N_BOX], y2 = lp[(size_t)3 * N_BOX];
  obox[tid * 4 + 0] = x1; obox[tid * 4 + 1] = y1;
  obox[tid * 4 + 2] = x2; obox[tid * 4 + 3] = y2;

  /* jnp.max(top_boxes) over all 1024x4 selected values */
  red[tid] = fmaxf(fmaxf(x1, y1), fmaxf(x2, y2));
  __syncthreads();
  for (int s = 512; s > 0; s >>= 1) {
    if (tid < s) red[tid] = fmaxf(red[tid], red[tid + s]);
    __syncthreads();
  }
  float off = (float)cls * (float)img * (red[0] + 1.0f);
  float sx1 = x1 + off, sy1 = y1 + off, sx2 = x2 + off, sy2 = y2 + off;
  sb[tid * 4 + 0] = sx1; sb[tid * 4 + 1] = sy1;
  sb[tid * 4 + 2] = sx2; sb[tid * 4 + 3] = sy2;
  area[tid] = (sx2 - sx1) * (sy2 - sy1);     /* from shifted boxes, as reference */
  score_o[tid] = score;
  cls_o[tid] = cls;
  unsigned long long bal = __ballot(score > 0.0f);
  if ((tid & 31) == 0) keepInit[tid >> 5] = (u32)bal;
}

/* ============ Kernel: suppression bitmask, one wave per 16x16 tile ============
 * Pairwise union pre-term area_i + area_j is the rank-2 product [a_i,1]·[1,a_j]^T,
 * computed on the matrix pipe with V_WMMA_F32_16X16X4_F32 (f32-exact).
 * Lane->(M,N) mapping matches the WMMA C/D layout, so acc[v] aligns with each
 * lane's inter term; __ballot packs two 16-bit row segments per step.
 */
__global__ void k_supmat(const float* __restrict__ sb, const float* __restrict__ area,
                         unsigned short* __restrict__ supH) {
  int w = blockIdx.x * (blockDim.x >> 5) + (threadIdx.x >> 5);  /* 0..4095 */
  int lane = threadIdx.x & 31;
  int ti = w >> 6, tj = w & 63;
  int half = lane >> 4, l15 = lane & 15;
  int j = tj * 16 + l15;

  /* A 16x4: lanes 0-15 hold K0,K1 = (area_row, 1); lanes 16-31 hold K2,K3 = 0.
   * B 4x16: lanes 0-15 hold rows K0,K1 = (1, area_col); lanes 16-31 = 0.     */
  v2f afrag, bfrag;
  afrag.x = (half == 0) ? area[ti * 16 + l15] : 0.0f;
  afrag.y = (half == 0) ? 1.0f : 0.0f;
  bfrag.x = (half == 0) ? 1.0f : 0.0f;
  bfrag.y = (half == 0) ? area[j] : 0.0f;
  v8f acc = {};
  acc = __builtin_amdgcn_wmma_f32_16x16x4_f32(false, afrag, false, bfrag,
                                              (short)0, acc, false, false);

  float jx1 = sb[j * 4 + 0], jy1 = sb[j * 4 + 1];
  float jx2 = sb[j * 4 + 2], jy2 = sb[j * 4 + 3];
#pragma unroll
  for (int v = 0; v < 8; ++v) {
    int i = ti * 16 + v + 8 * half;       /* C/D layout: M = v (+8 for hi half) */
    float ix1 = sb[i * 4 + 0], iy1 = sb[i * 4 + 1];
    float ix2 = sb[i * 4 + 2], iy2 = sb[i * 4 + 3];
    float iw = fmaxf(fminf(ix2, jx2) - fmaxf(ix1, jx1), 0.0f);
    float ih = fmaxf(fminf(iy2, jy2) - fmaxf(iy1, jy1), 0.0f);
    float inter = iw * ih;
    float uni = acc[v] - inter;           /* (area_i + area_j) - inter */
    float iou = (uni > 0.0f) ? (inter / uni) : 0.0f;
    bool sup = (iou > IOU_THR) && (j > i);
    unsigned bal = (unsigned)__ballot(sup);
    if (lane == 0)  supH[(size_t)(ti * 16 + v) * 64 + tj]     = (unsigned short)(bal & 0xFFFFu);
    if (lane == 16) supH[(size_t)(ti * 16 + v + 8) * 64 + tj] = (unsigned short)(bal >> 16);
  }
}

/* ============ Kernel: greedy NMS scan (1 wave) + masked outputs ============ */
__global__ void __launch_bounds__(1024)
k_nms_out(const float* __restrict__ obox, const float* __restrict__ score,
          const int* __restrict__ cls, const u32* __restrict__ keepInit,
          const u32* __restrict__ sup32, float* __restrict__ out) {
  __shared__ u32 keepW[32];
  int tid = threadIdx.x;
  if (tid < 32) {                      /* wave 0: 1024-bit keep mask, 1 u32/lane */
    int lane = tid;
    u32 keep = keepInit[lane];
    for (int i = 0; i < 1024; ++i) {
      u32 kw = __shfl(keep, i >> 5, 32);
      if ((kw >> (i & 31)) & 1u)       /* row i survives -> suppress j>i row */
        keep &= ~sup32[(size_t)i * 32 + lane];
    }
    keepW[lane] = keep;
  }
  __syncthreads();
  u32 k = (keepW[tid >> 5] >> (tid & 31)) & 1u;
  float fk = (float)k;
  out[tid * 4 + 0] = obox[tid * 4 + 0] * fk;
  out[tid * 4 + 1] = obox[tid * 4 + 1] * fk;
  out[tid * 4 + 2] = obox[tid * 4 + 2] * fk;
  out[tid * 4 + 3] = obox[tid * 4 + 3] * fk;
  out[4096 + tid] = score[tid] * fk;
  out[5120 + tid] = k ? (float)cls[tid] : 0.0f;
  out[6144 + tid] = fk;
}

/* ============ host-side launcher ============ */
extern "C" void kernel_launch(void* const* d_in, const int* in_sizes, int n_in,
                              void* d_out, int out_size, void* d_ws, size_t ws_size,
                              hipStream_t stream) {
  const float* locs = (const float*)d_in[0];
  const float* labels = (const float*)d_in[1];
  char* ws = (char*)d_ws;
  u32* counter = (u32*)(ws + WS_COUNTER);
  u64* cand = (u64*)(ws + WS_CAND);
  u64* l1out = (u64*)(ws + WS_L1OUT);
  u64* topk = (u64*)(ws + WS_TOPK);
  float* sb = (float*)(ws + WS_SB);
  float* obox = (float*)(ws + WS_OBOX);
  float* area = (float*)(ws + WS_AREA);
  float* score = (float*)(ws + WS_SCORE);
  int* cls = (int*)(ws + WS_CLS);
  u32* keepInit = (u32*)(ws + WS_KEEPINIT);
  unsigned short* supH = (unsigned short*)(ws + WS_SUP);

  hipMemsetAsync(counter, 0, sizeof(u32), stream);
  k_softmax_cand<<<(BN + 255) / 256, 256, 0, stream>>>(labels, counter, cand);
  k_topk_l1<<<64, 1024, 0, stream>>>(counter, cand, l1out);
  k_topk_l2<<<1, 1024, 0, stream>>>(l1out, topk);
  k_decode<<<1, 1024, 0, stream>>>(locs, topk, sb, obox, area, score, cls, keepInit);
  k_supmat<<<512, 256, 0, stream>>>(sb, area, supH);
  k_nms_out<<<1, 1024, 0, stream>>>(obox, score, cls, keepInit, (const u32*)supH,
                                    (float*)d_out);
}